// clip_only_hd_loss_45131516347004
// MI455X (gfx1250) — compile-verified
//
#include <hip/hip_runtime.h>
#include <math.h>

// ---------------- static problem config (mirrors reference) ----------------
#define Qn 4096      // number of queries
#define Vn 2048      // number of videos
#define Dn 8192      // embedding dim
// loss constants
#define MARGIN   0.2f
#define ALPHA    32.0f
#define MRG      0.1f
#define LF_NCE   0.02f
#define LF_QDL   0.04f

#define NEG_INF (-__builtin_inff())

typedef float v2f __attribute__((ext_vector_type(2)));
typedef float v8f __attribute__((ext_vector_type(8)));

// ---------------- workspace layout (float offsets) ----------------
// scalars: [0]=nce_t2v_sum [1]=nce_v2t_sum [2]=trip_t2v_sum [3]=trip_v2t_sum [4]=qdl_sum
#define WS_COL_M    64                      // 8 segs * Vn  (column LSE running max, scores_)
#define WS_COL_S    (WS_COL_M + 8*Vn)       // 8 segs * Vn  (column LSE running sum, scores_)
#define WS_TRIP_MX  (WS_COL_S + 8*Vn)       // 8 segs * Vn  (column neg-max, scores)
#define WS_EO       (WS_TRIP_MX + 8*Vn)     // Vn pair dots  even.odd
#define WS_EE       (WS_EO + Vn)            // Vn            even norms^2
#define WS_OO       (WS_EE + Vn)            // Vn            odd norms^2
#define WS_TOTAL    (WS_OO + Vn)            // = 55360 floats (~216 KB)

// ---------------- helpers ----------------
__device__ __forceinline__ void lse_add(float& m, float& s, float x) {
  if (x > m) { s = s * __expf(m - x) + 1.0f; m = x; }
  else       { s += __expf(x - m); }
}
__device__ __forceinline__ void lse_merge(float& m, float& s, float m2, float s2) {
  float M = fmaxf(m, m2);
  s = s * __expf(m - M) + s2 * __expf(m2 - M);
  m = M;
}
__device__ __forceinline__ float sel8(v8f v, int r) {
  switch (r) {
    case 0: return v[0]; case 1: return v[1]; case 2: return v[2]; case 3: return v[3];
    case 4: return v[4]; case 5: return v[5]; case 6: return v[6];
  }
  return v[7];
}

// ---------------- kernels ----------------
__global__ void hdx_init(float* ws) {
  int i = blockIdx.x * blockDim.x + threadIdx.x;
  if (i < WS_TOTAL) ws[i] = 0.0f;
}

// one wave32 per query row: t2v part of clip_nce (logsumexp over V minus pos)
__global__ __launch_bounds__(256) void hdx_nce_rows(const float* __restrict__ S_, float* ws) {
  int gid  = blockIdx.x * blockDim.x + threadIdx.x;
  int q    = gid >> 5;
  int lane = threadIdx.x & 31;
  const float4* row = (const float4*)(S_ + (size_t)q * Vn);
  float m = NEG_INF, s = 0.0f;
  #pragma unroll 4
  for (int i = lane; i < Vn / 4; i += 32) {
    float4 x = row[i];
    lse_add(m, s, x.x); lse_add(m, s, x.y); lse_add(m, s, x.z); lse_add(m, s, x.w);
  }
  #pragma unroll
  for (int off = 16; off; off >>= 1) {
    float m2 = __shfl_xor(m, off);
    float s2 = __shfl_xor(s, off);
    lse_merge(m, s, m2, s2);
  }
  if (lane == 0) {
    float lse = m + logf(s);
    float pos = S_[(size_t)q * Vn + (q >> 1)];
    atomicAdd(&ws[0], lse - pos);
  }
}

// column scans, split into 8 row-segments: online-LSE over scores_ columns (v2t nce)
// and neg-max over scores columns (v2t triplet, excluding the 2 positive rows)
__global__ __launch_bounds__(256) void hdx_cols_partial(const float* __restrict__ S_,
                                                        const float* __restrict__ S,
                                                        float* ws) {
  int t   = blockIdx.x * blockDim.x + threadIdx.x;   // 0 .. 8*Vn-1
  int c   = t & (Vn - 1);
  int seg = t >> 11;                                  // Vn==2048
  int r0  = seg * (Qn / 8);
  int p0  = 2 * c, p1 = 2 * c + 1;
  float m = NEG_INF, s = 0.0f, mx = NEG_INF;
  #pragma unroll 4
  for (int r = r0; r < r0 + Qn / 8; ++r) {
    float xn = S_[(size_t)r * Vn + c];
    lse_add(m, s, xn);
    float xt = S[(size_t)r * Vn + c];
    if (r != p0 && r != p1) mx = fmaxf(mx, xt);
  }
  ws[WS_COL_M   + t] = m;
  ws[WS_COL_S   + t] = s;
  ws[WS_TRIP_MX + t] = mx;
}

__global__ __launch_bounds__(256) void hdx_cols_finish(const float* __restrict__ S_,
                                                       const float* __restrict__ S,
                                                       float* ws) {
  int c = blockIdx.x * blockDim.x + threadIdx.x;     // 0..Vn-1
  float m = NEG_INF, s = 0.0f, mx = NEG_INF;
  #pragma unroll
  for (int seg = 0; seg < 8; ++seg) {
    int t = seg * Vn + c;
    lse_merge(m, s, ws[WS_COL_M + t], ws[WS_COL_S + t]);
    mx = fmaxf(mx, ws[WS_TRIP_MX + t]);
  }
  float lse = m + logf(s);
  // group logsumexp over the column's own two queries
  float a  = S_[(size_t)(2 * c) * Vn + c];
  float b  = S_[(size_t)(2 * c + 1) * Vn + c];
  float mg = fmaxf(a, b);
  float l2 = mg + logf(__expf(a - mg) + __expf(b - mg));
  atomicAdd(&ws[1], lse - l2);
  // triplet v2t hinge
  float pa = S[(size_t)(2 * c) * Vn + c];
  float pb = S[(size_t)(2 * c + 1) * Vn + c];
  float tl = fmaxf(MARGIN + mx - 0.5f * (pa + pb), 0.0f);
  atomicAdd(&ws[3], tl);
}

// one block per query row: rank-rand_idx[q] largest negative via iterative max extraction
__global__ __launch_bounds__(256) void hdx_trip_rows(const float* __restrict__ S,
                                                     const int* __restrict__ rnd,
                                                     float* ws) {
  __shared__ float row[Vn];
  __shared__ float rv[256];
  __shared__ int   ri[256];
  int q = blockIdx.x, tid = threadIdx.x;
  int posc = q >> 1;
  for (int c = tid; c < Vn; c += 256) row[c] = S[(size_t)q * Vn + c];
  __syncthreads();
  if (tid == 0) row[posc] = NEG_INF;  // exclude positive column
  __syncthreads();
  int samp = rnd[q];                  // 0-based rank among negatives
  float neg = 0.0f;
  for (int it = 0; it <= samp; ++it) {
    float bm = NEG_INF; int bi = 0;
    for (int c = tid; c < Vn; c += 256) {
      float x = row[c];
      if (x > bm) { bm = x; bi = c; }
    }
    rv[tid] = bm; ri[tid] = bi;
    __syncthreads();
    for (int s = 128; s > 0; s >>= 1) {
      if (tid < s && rv[tid + s] > rv[tid]) { rv[tid] = rv[tid + s]; ri[tid] = ri[tid + s]; }
      __syncthreads();
    }
    if (tid == 0) {
      if (it == samp) neg = rv[0];
      else            row[ri[0]] = NEG_INF;
    }
    __syncthreads();
  }
  if (tid == 0) {
    float pos = S[(size_t)q * Vn + posc];
    atomicAdd(&ws[2], fmaxf(MARGIN + neg - pos, 0.0f));
  }
}

// qdl pair dots & norms via V_WMMA_F32_16X16X4_F32.
// wave w: tile = w>>3 (16 pairs), seg = w&7 (K segment of Dn/8 floats).
// A(16x4) per-lane layout: lanes 0-15 hold row m=lane cols k..k+1, lanes 16-31 row m=lane-16
// cols k+2..k+3 -> the same float2 registers also serve as the B-slice of the transpose.
__global__ __launch_bounds__(256) void hdx_qdl_wmma(const float* __restrict__ X, float* ws) {
  int wid  = (blockIdx.x * blockDim.x + threadIdx.x) >> 5;
  int lane = threadIdx.x & 31;
  int tile = wid >> 3;
  int seg  = wid & 7;
  int mrow = lane & 15;
  int half = lane >> 4;
  size_t erow = (size_t)(tile * 32 + 2 * mrow);        // even query row
  const float* ep = X + erow * Dn + 2 * half;
  const float* op = ep + Dn;                           // odd partner row
  int k0 = seg * (Dn / 8);
  v8f aeo = {}; v8f aee = {}; v8f aoo = {};
  #pragma unroll 4
  for (int k = k0; k < k0 + Dn / 8; k += 4) {
    v2f ev = *(const v2f*)(ep + k);
    v2f ov = *(const v2f*)(op + k);
    aeo = __builtin_amdgcn_wmma_f32_16x16x4_f32(false, ev, false, ov, (short)0, aeo, false, false);
    aee = __builtin_amdgcn_wmma_f32_16x16x4_f32(false, ev, false, ev, (short)0, aee, false, false);
    aoo = __builtin_amdgcn_wmma_f32_16x16x4_f32(false, ov, false, ov, (short)0, aoo, false, false);
  }
  // diagonal extraction: (i,i) i<8 -> lane i, reg i ; i>=8 -> lane i+16, reg i-8
  int di = (lane < 8) ? lane : ((lane >= 24) ? (lane - 16) : -1);
  int r  = (lane < 8) ? lane : (lane - 24);
  if (di >= 0) {
    int pair = tile * 16 + di;
    atomicAdd(&ws[WS_EO + pair], sel8(aeo, r));
    atomicAdd(&ws[WS_EE + pair], sel8(aee, r));
    atomicAdd(&ws[WS_OO + pair], sel8(aoo, r));
  }
}

__global__ __launch_bounds__(256) void hdx_qdl_finish(float* ws) {
  int p = blockIdx.x * blockDim.x + threadIdx.x;       // 0..Vn-1 pairs
  float dot = ws[WS_EO + p];
  float ne  = fmaxf(sqrtf(ws[WS_EE + p]), 1e-12f);
  float no  = fmaxf(sqrtf(ws[WS_OO + p]), 1e-12f);
  float cosv = dot / (ne * no);
  float z = ALPHA * (cosv + MRG);
  float l = (z > 0.0f) ? (z + log1pf(__expf(-z))) : log1pf(__expf(z));
  atomicAdd(&ws[4], 2.0f * l);                         // both queries of the pair
}

__global__ void hdx_final(const float* __restrict__ ws, float* out) {
  out[0] = LF_NCE * (ws[0] / (float)Qn + ws[1] / (float)Vn)
         + ws[2] / (float)Qn + ws[3] / (float)Vn
         + LF_QDL * (ws[4] / (float)Qn);
}

// ---------------- launch ----------------
extern "C" void kernel_launch(void* const* d_in, const int* in_sizes, int n_in,
                              void* d_out, int out_size, void* d_ws, size_t ws_size,
                              hipStream_t stream) {
  (void)in_sizes; (void)n_in; (void)out_size; (void)ws_size;
  const float* S   = (const float*)d_in[0];  // hd_similarity_scores   [Q,V] (triplet)
  const float* S_  = (const float*)d_in[1];  // hd_similarity_scores_  [Q,V] (nce)
  const float* X   = (const float*)d_in[2];  // query_hd               [Q,D]
  const int*   rnd = (const int*)d_in[4];    // rand_idx               [Q]
  float* ws  = (float*)d_ws;
  float* out = (float*)d_out;

  hdx_init        <<<(WS_TOTAL + 255) / 256, 256, 0, stream>>>(ws);
  hdx_nce_rows    <<<Qn / 8, 256, 0, stream>>>(S_, ws);        // 4096 waves
  hdx_cols_partial<<<(8 * Vn) / 256, 256, 0, stream>>>(S_, S, ws);
  hdx_trip_rows   <<<Qn, 256, 0, stream>>>(S, rnd, ws);
  hdx_qdl_wmma    <<<128, 256, 0, stream>>>(X, ws);            // 1024 waves, 3 WMMA/step
  hdx_cols_finish <<<Vn / 256, 256, 0, stream>>>(S_, S, ws);
  hdx_qdl_finish  <<<Vn / 256, 256, 0, stream>>>(ws);
  hdx_final       <<<1, 1, 0, stream>>>(ws, out);
}